// pRNN_5050881540306
// MI455X (gfx1250) — compile-verified
//
#include <hip/hip_runtime.h>

typedef float v2f __attribute__((ext_vector_type(2)));
typedef float v4f __attribute__((ext_vector_type(4)));
typedef float v8f __attribute__((ext_vector_type(8)));

constexpr int Bsz = 16384;  // batch
constexpr int Isz = 512;    // input trace width
constexpr int Hsz = 256;    // hidden width

// Block = 128 threads = 4 waves, all owning ONE 16-row batch tile. Every wave
// redundantly evaluates the serial node chain (4 WMMAs/node via the packed-
// column trick), then owns 4 of node 7's 16 output column-tiles. Redundant
// gathers hit the same L2 lines, so HBM traffic is unchanged while resident
// wave count doubles for latency hiding.
__global__ __launch_bounds__(128) void prnn_wmma_kernel(
    const float* __restrict__ x,      // (B, I)
    const float* __restrict__ h,      // (8, B, H)
    const float* __restrict__ conv_w, // (I,)
    const float* __restrict__ conv_b, // (I,)
    const float* __restrict__ W,      // (7, H, 16)
    const float* __restrict__ bia,    // (7, H)
    float* __restrict__ out)          // (B, H)
{
    // Intermediate nodes only feed cols 32*j (j<8) downstream; stage those 8
    // weight rows per node, zero-padded to 16 so all lanes load uniformly.
    __shared__ float lds_wp[6][16][16];  // [i][j][k] = W[i][32*j][k], j>=8 -> 0
    __shared__ float lds_w7[256][16];    // full W[6]
    __shared__ float lds_bp[6][16];      // b[i][32*j], j>=8 -> 0
    __shared__ float lds_b7[256];        // b[6][:]
    // Trace scratch, TRANSPOSED: [wave][col j][row]. A lane's 8 D values
    // (same col, rows kh*8..kh*8+7) are contiguous -> 2x ds_store_b128.
    __shared__ float lds_fwd[4][16][16];

    const int tid  = threadIdx.x;
    const int wave = tid >> 5;
    const int lane = tid & 31;
    const int kh   = lane >> 4;          // K-half select: lanes 16-31 hold K+2
    const int m    = lane & 15;          // A: row within tile; B/D: column
    const int tileIdx = blockIdx.x;
    const long grow = (long)tileIdx * 16 + m;  // global batch row for gathers

    // ---- 1) Issue ALL global gathers first (register-only, independent of
    // LDS staging): their latency overlaps staging + the serial WMMA chain.
    // A operand for V_WMMA_F32_16X16X4_F32 (16x4 f32 = 2 VGPRs/lane):
    //   chunk c covers K = c*4 + kh*2 + {0,1};  g = [8 fwd | 8 recurrent].
    v2f Arec[7][2];
    #pragma unroll
    for (int i = 1; i <= 7; ++i) {
        const int jr = (i < 7) ? i + 1 : 7;   // JR = min(i+1, 7)
        #pragma unroll
        for (int cc = 0; cc < 2; ++cc) {
            #pragma unroll
            for (int v = 0; v < 2; ++v) {
                const int j = cc * 4 + kh * 2 + v;   // 0..7
                Arec[i - 1][cc][v] =
                    h[(long)jr * Bsz * Hsz + grow * Hsz + 32 * j + 16];
            }
        }
    }
    float xv[2][2], cwv[2][2], cbv[2][2];
    #pragma unroll
    for (int c = 0; c < 2; ++c) {
        #pragma unroll
        for (int v = 0; v < 2; ++v) {
            const int col = 64 * (c * 4 + kh * 2 + v);
            xv[c][v]  = x[grow * Isz + col];
            cwv[c][v] = conv_w[col];
            cbv[c][v] = conv_b[col];
        }
    }

    // ---- 2) Stage weights/biases into LDS ----
    for (int s = tid; s < 6 * 16 * 16; s += 128) {
        const int i = s >> 8, j = (s >> 4) & 15, k = s & 15;
        lds_wp[i][j][k] = (j < 8) ? W[(i * Hsz + 32 * j) * 16 + k] : 0.f;
    }
    for (int s = tid; s < Hsz * 16; s += 128)
        (&lds_w7[0][0])[s] = W[6 * Hsz * 16 + s];
    if (tid < 96) {
        const int i = tid >> 4, j = tid & 15;
        lds_bp[i][j] = (j < 8) ? bia[i * Hsz + 32 * j] : 0.f;
    }
    for (int s = tid; s < Hsz; s += 128)
        lds_b7[s] = bia[6 * Hsz + s];
    __syncthreads();

    // ---- 3) Node 1 forward inputs: relu(x*cw+cb) at cols 64*k ----
    v2f A0, A1;
    #pragma unroll
    for (int v = 0; v < 2; ++v) {
        A0[v] = fmaxf(fmaf(xv[0][v], cwv[0][v], cbv[0][v]), 0.f);
        A1[v] = fmaxf(fmaf(xv[1][v], cwv[1][v], cbv[1][v]), 0.f);
    }

    // ---- 4) Serial chain: nodes 1..6 (packed single-tile WMMA each) ----
    #pragma unroll
    for (int i = 1; i <= 6; ++i) {
        v8f acc = {0.f, 0.f, 0.f, 0.f, 0.f, 0.f, 0.f, 0.f};
        #pragma unroll
        for (int c = 0; c < 4; ++c) {
            const v2f Bop = *(const v2f*)&lds_wp[i - 1][m][c * 4 + kh * 2];
            const v2f Aop = (c == 0) ? A0 : (c == 1) ? A1 : Arec[i - 1][c - 2];
            acc = __builtin_amdgcn_wmma_f32_16x16x4_f32(
                false, Aop, false, Bop, (short)0, acc, false, false);
        }
        const float bv = lds_bp[i - 1][m];
        // D layout: VGPR r = (row kh*8+r, col m). Transposed scratch makes
        // this lane's 8 values contiguous: two b128 stores.
        v4f lo, hi;
        #pragma unroll
        for (int r = 0; r < 4; ++r) {
            lo[r] = fmaxf(acc[r]     + bv, 0.f);
            hi[r] = fmaxf(acc[r + 4] + bv, 0.f);
        }
        *(v4f*)&lds_fwd[wave][m][kh * 8]     = lo;
        *(v4f*)&lds_fwd[wave][m][kh * 8 + 4] = hi;
        // Next node's forward inputs (same wave; hw dscnt wait via compiler):
        // element (row m, trace col k) lives at lds_fwd[wave][k][m].
        A0[0] = lds_fwd[wave][kh * 2    ][m];
        A0[1] = lds_fwd[wave][kh * 2 + 1][m];
        A1[0] = lds_fwd[wave][kh * 2 + 4][m];
        A1[1] = lds_fwd[wave][kh * 2 + 5][m];
    }

    // ---- 5) Node 7: full 256-col output; this wave owns 4 of 16 tiles ----
    const long rowbase = (long)tileIdx * 16 + kh * 8;
    #pragma unroll
    for (int tt = 0; tt < 4; ++tt) {
        const int t = wave * 4 + tt;
        const int n = t * 16 + m;
        v8f acc = {0.f, 0.f, 0.f, 0.f, 0.f, 0.f, 0.f, 0.f};
        #pragma unroll
        for (int c = 0; c < 4; ++c) {
            const v2f Bop = *(const v2f*)&lds_w7[n][c * 4 + kh * 2];
            const v2f Aop = (c == 0) ? A0 : (c == 1) ? A1 : Arec[6][c - 2];
            acc = __builtin_amdgcn_wmma_f32_16x16x4_f32(
                false, Aop, false, Bop, (short)0, acc, false, false);
        }
        const float bv = lds_b7[n];
        #pragma unroll
        for (int r = 0; r < 8; ++r)
            out[(rowbase + r) * Hsz + n] = fmaxf(acc[r] + bv, 0.f);
    }
}

extern "C" void kernel_launch(void* const* d_in, const int* in_sizes, int n_in,
                              void* d_out, int out_size, void* d_ws, size_t ws_size,
                              hipStream_t stream) {
    (void)in_sizes; (void)n_in; (void)out_size; (void)d_ws; (void)ws_size;
    const float* x      = (const float*)d_in[0];
    const float* h      = (const float*)d_in[1];
    const float* conv_w = (const float*)d_in[2];
    const float* conv_b = (const float*)d_in[3];
    // d_in[4] = W0, d_in[5] = b0: dead in the reference dataflow (traces[0]
    // is never consumed — node 1 reads trace_in, nodes 2..7 read traces[i-1]).
    const float* W      = (const float*)d_in[6];
    const float* bia    = (const float*)d_in[7];
    float* out = (float*)d_out;

    dim3 grid(Bsz / 16);   // one 16-row batch tile per 128-thread block
    dim3 block(128);
    hipLaunchKernelGGL(prnn_wmma_kernel, grid, block, 0, stream,
                       x, h, conv_w, conv_b, W, bia, out);
}